// EEGGraphConvNetDeep_61409442398714
// MI455X (gfx1250) — compile-verified
//
#include <hip/hip_runtime.h>
#include <hip/hip_bf16.h>
#include <math.h>

#define N_NODES  100000
#define N_EDGES  1600000
#define N_GRAPHS 256

typedef __attribute__((ext_vector_type(2))) float v2f;
typedef __attribute__((ext_vector_type(8))) float v8f;

// ---------------------------------------------------------------------------
// Zero fill
// ---------------------------------------------------------------------------
__global__ void zero_f32(float* __restrict__ p, int n) {
    int i = blockIdx.x * blockDim.x + threadIdx.x;
    if (i < n) p[i] = 0.0f;
}

// ---------------------------------------------------------------------------
// WMMA f32 GEMM:  out[n_rows x N] = act(in + bias_prev) @ W
//   K    : inner dim (multiple of 4)
//   NT   : number of 16-wide output tiles (N = NT*16)
//   ACT  : apply (x + bias_prev[k]) then leaky-relu(0.01) to A elements
//   W is [K x wN] row-major; columns >= wN are zero-padded in LDS.
// One wave computes one 16-row tile across all NT column tiles.
// ---------------------------------------------------------------------------
template <int K, int NT, bool ACT>
__global__ __launch_bounds__(128)
void gcn_gemm_wmma(const float* __restrict__ in, int ldin,
                   const float* __restrict__ W, int wN,
                   const float* __restrict__ bias_prev,
                   float* __restrict__ out, int ldout,
                   int n_rows)
{
    constexpr int N = NT * 16;
    __shared__ float Wlds[K * N];

    for (int i = threadIdx.x; i < K * N; i += blockDim.x) {
        int k = i / N, n = i % N;
        Wlds[i] = (n < wN) ? W[k * wN + n] : 0.0f;
    }
    __syncthreads();

    const int wave = threadIdx.x >> 5;            // 4 waves / block
    const int lane = threadIdx.x & 31;
    const int rowTile = blockIdx.x * 4 + wave;    // wave-uniform guard keeps EXEC full
    if (rowTile * 16 >= n_rows) return;

    const int row   = rowTile * 16 + (lane & 15);
    const int kHalf = (lane >> 4) * 2;            // lanes 0-15 -> K{0,1}, 16-31 -> K{2,3}
    const int nCol  = lane & 15;

    v8f acc[NT];
    const v8f vzero = {0.f, 0.f, 0.f, 0.f, 0.f, 0.f, 0.f, 0.f};
#pragma unroll
    for (int t = 0; t < NT; ++t) acc[t] = vzero;

    const float* rowp = in + (size_t)row * ldin;
    for (int k0 = 0; k0 < K; k0 += 4) {
        const int ka = k0 + kHalf;
        float a0 = rowp[ka];
        float a1 = rowp[ka + 1];
        if constexpr (ACT) {
            a0 += bias_prev[ka];     a0 = (a0 > 0.f) ? a0 : 0.01f * a0;
            a1 += bias_prev[ka + 1]; a1 = (a1 > 0.f) ? a1 : 0.01f * a1;
        }
        v2f a = {a0, a1};
#pragma unroll
        for (int t = 0; t < NT; ++t) {
            const int n = t * 16 + nCol;
            v2f b = {Wlds[ka * N + n], Wlds[(ka + 1) * N + n]};
            acc[t] = __builtin_amdgcn_wmma_f32_16x16x4_f32(
                false, a, false, b, (short)0, acc[t], false, false);
        }
    }

    // C/D layout: VGPR r -> M = r + 8*(lane>=16), N = lane&15
    const int mBase = rowTile * 16 + ((lane >> 4) << 3);
#pragma unroll
    for (int t = 0; t < NT; ++t) {
#pragma unroll
        for (int r = 0; r < 8; ++r) {
            out[(size_t)(mBase + r) * ldout + t * 16 + nCol] = acc[t][r];
        }
    }
}

// ---------------------------------------------------------------------------
// Edge scatter-add: agg[dst] += h[src]   (float4 path, D % 4 == 0)
// ---------------------------------------------------------------------------
template <int D>
__global__ void scatter_add_vec4(const float* __restrict__ h, int ldh,
                                 const int* __restrict__ esrc,
                                 const int* __restrict__ edst,
                                 float* __restrict__ agg, int ldagg)
{
    constexpr int G = D / 4;
    int tid = blockIdx.x * blockDim.x + threadIdx.x;
    if (tid >= N_EDGES * G) return;
    int e = tid / G, g = tid % G;
    int s = esrc[e], d = edst[e];
    const float4 v = *reinterpret_cast<const float4*>(h + (size_t)s * ldh + g * 4);
    float* ap = agg + (size_t)d * ldagg + g * 4;
    atomicAdd(ap + 0, v.x);
    atomicAdd(ap + 1, v.y);
    atomicAdd(ap + 2, v.z);
    atomicAdd(ap + 3, v.w);
}

// Scalar scatter for D = 50 (h stored with stride ldh=64, agg stride 50)
__global__ void scatter_add_50(const float* __restrict__ h,
                               const int* __restrict__ esrc,
                               const int* __restrict__ edst,
                               float* __restrict__ agg)
{
    int tid = blockIdx.x * blockDim.x + threadIdx.x;
    if (tid >= N_EDGES * 50) return;
    int e = tid / 50, f = tid % 50;
    int s = esrc[e], d = edst[e];
    atomicAdd(agg + (size_t)d * 50 + f, h[(size_t)s * 64 + f]);
}

// ---------------------------------------------------------------------------
// BatchNorm statistics over nodes (feature dim 50): sum and sum of squares
// ---------------------------------------------------------------------------
__global__ __launch_bounds__(256)
void bn_stats(const float* __restrict__ agg,
              float* __restrict__ sums, float* __restrict__ sumsq)
{
    __shared__ float ls[50], lq[50];
    if (threadIdx.x < 50) { ls[threadIdx.x] = 0.f; lq[threadIdx.x] = 0.f; }
    __syncthreads();
    const int total = N_NODES * 50;
    for (int i = blockIdx.x * blockDim.x + threadIdx.x; i < total;
         i += gridDim.x * blockDim.x) {
        float v = agg[i];
        int f = i % 50;
        atomicAdd(&ls[f], v);
        atomicAdd(&lq[f], v * v);
    }
    __syncthreads();
    if (threadIdx.x < 50) {
        atomicAdd(&sums[threadIdx.x],  ls[threadIdx.x]);
        atomicAdd(&sumsq[threadIdx.x], lq[threadIdx.x]);
    }
}

// ---------------------------------------------------------------------------
// BN apply + leaky-relu + global_add_pool into per-graph accumulators
// ---------------------------------------------------------------------------
__global__ void bn_pool(const float* __restrict__ agg,
                        const int* __restrict__ batch,
                        const float* __restrict__ sums,
                        const float* __restrict__ sumsq,
                        const float* __restrict__ gamma,
                        const float* __restrict__ beta,
                        float* __restrict__ pool)
{
    int tid = blockIdx.x * blockDim.x + threadIdx.x;
    if (tid >= N_NODES * 50) return;
    int node = tid / 50, f = tid % 50;
    const float invN = 1.0f / (float)N_NODES;
    float mu  = sums[f] * invN;
    float var = sumsq[f] * invN - mu * mu;
    float inv = rsqrtf(var + 1e-5f);
    float v = (agg[tid] - mu) * inv * gamma[f] + beta[f];
    v = (v > 0.f) ? v : 0.01f * v;
    atomicAdd(&pool[(size_t)batch[node] * 50 + f], v);
}

// ---------------------------------------------------------------------------
// MLP head: 50 -> 30 -> 20 -> 2 (+log_softmax). One thread per graph.
// ---------------------------------------------------------------------------
__global__ __launch_bounds__(256)
void head_kernel(const float* __restrict__ pool,
                 const float* __restrict__ fc1W, const float* __restrict__ fc1b,
                 const float* __restrict__ fc2W, const float* __restrict__ fc2b,
                 const float* __restrict__ fc3W, const float* __restrict__ fc3b,
                 float* __restrict__ out)
{
    __shared__ float w1[50 * 30], w2[30 * 20], w3[20 * 2];
    __shared__ float b1[30], b2[20], b3[2];
    for (int i = threadIdx.x; i < 1500; i += 256) w1[i] = fc1W[i];
    for (int i = threadIdx.x; i < 600;  i += 256) w2[i] = fc2W[i];
    if (threadIdx.x < 40) w3[threadIdx.x] = fc3W[threadIdx.x];
    if (threadIdx.x < 30) b1[threadIdx.x] = fc1b[threadIdx.x];
    if (threadIdx.x < 20) b2[threadIdx.x] = fc2b[threadIdx.x];
    if (threadIdx.x < 2)  b3[threadIdx.x] = fc3b[threadIdx.x];
    __syncthreads();

    int g = blockIdx.x * blockDim.x + threadIdx.x;
    if (g >= N_GRAPHS) return;

    float gin[50];
#pragma unroll
    for (int i = 0; i < 50; ++i) gin[i] = pool[g * 50 + i];

    float a1[30];
    for (int j = 0; j < 30; ++j) {
        float s = b1[j];
        for (int i = 0; i < 50; ++i) s += gin[i] * w1[i * 30 + j];
        a1[j] = (s > 0.f) ? s : 0.01f * s;
    }
    float a2[20];
    for (int j = 0; j < 20; ++j) {
        float s = b2[j];
        for (int i = 0; i < 30; ++i) s += a1[i] * w2[i * 20 + j];
        a2[j] = (s > 0.f) ? s : 0.01f * s;
    }
    float z0 = b3[0], z1 = b3[1];
    for (int i = 0; i < 20; ++i) {
        z0 += a2[i] * w3[i * 2 + 0];
        z1 += a2[i] * w3[i * 2 + 1];
    }
    float m   = fmaxf(z0, z1);
    float lse = m + logf(expf(z0 - m) + expf(z1 - m));
    out[g * 2 + 0] = z0 - lse;
    out[g * 2 + 1] = z1 - lse;
}

// ---------------------------------------------------------------------------
// Launch
// ---------------------------------------------------------------------------
extern "C" void kernel_launch(void* const* d_in, const int* in_sizes, int n_in,
                              void* d_out, int out_size, void* d_ws, size_t ws_size,
                              hipStream_t stream)
{
    const float* x     = (const float*)d_in[0];
    const int*   ei    = (const int*)d_in[1];
    const int*   batch = (const int*)d_in[2];
    const float* W1 = (const float*)d_in[3];   const float* b1 = (const float*)d_in[4];
    const float* W2 = (const float*)d_in[5];   const float* b2 = (const float*)d_in[6];
    const float* W3 = (const float*)d_in[7];   const float* b3 = (const float*)d_in[8];
    const float* W4 = (const float*)d_in[9];   /* b4 cancels inside BatchNorm */
    const float* bn_g = (const float*)d_in[11];
    const float* bn_b = (const float*)d_in[12];
    const float* fc1W = (const float*)d_in[13]; const float* fc1b = (const float*)d_in[14];
    const float* fc2W = (const float*)d_in[15]; const float* fc2b = (const float*)d_in[16];
    const float* fc3W = (const float*)d_in[17]; const float* fc3b = (const float*)d_in[18];

    const int* esrc = ei;
    const int* edst = ei + N_EDGES;

    float* ws    = (float*)d_ws;
    float* bufA  = ws;                       // 100000*64 floats
    float* bufB  = ws + (size_t)N_NODES * 64;
    float* sums  = ws + (size_t)N_NODES * 128;
    float* sumsq = sums + 50;
    float* pool  = sumsq + 50;               // 256*50 floats

    const int rowTiles  = N_NODES / 16;                    // 6250
    const int gemmGrid  = (rowTiles + 3) / 4;              // 4 waves / block

    auto zero = [&](float* p, int n) {
        zero_f32<<<(n + 255) / 256, 256, 0, stream>>>(p, n);
    };

    // ---- Layer 1: 128 -> 16 --------------------------------------------
    zero(bufB, N_NODES * 16);
    gcn_gemm_wmma<128, 1, false><<<gemmGrid, 128, 0, stream>>>(
        x, 128, W1, 16, nullptr, bufA, 16, N_NODES);
    scatter_add_vec4<16><<<(N_EDGES * 4 + 255) / 256, 256, 0, stream>>>(
        bufA, 16, esrc, edst, bufB, 16);

    // ---- Layer 2: 16 -> 32 (fused bias1 + lrelu on A-load) -------------
    gcn_gemm_wmma<16, 2, true><<<gemmGrid, 128, 0, stream>>>(
        bufB, 16, W2, 32, b1, bufA, 32, N_NODES);
    zero(bufB, N_NODES * 32);
    scatter_add_vec4<32><<<(N_EDGES * 8 + 255) / 256, 256, 0, stream>>>(
        bufA, 32, esrc, edst, bufB, 32);

    // ---- Layer 3: 32 -> 64 ---------------------------------------------
    gcn_gemm_wmma<32, 4, true><<<gemmGrid, 128, 0, stream>>>(
        bufB, 32, W3, 64, b2, bufA, 64, N_NODES);
    zero(bufB, N_NODES * 64);
    scatter_add_vec4<64><<<(N_EDGES * 16 + 255) / 256, 256, 0, stream>>>(
        bufA, 64, esrc, edst, bufB, 64);

    // ---- Layer 4: 64 -> 50 (W padded to 64 cols in LDS) ----------------
    gcn_gemm_wmma<64, 4, true><<<gemmGrid, 128, 0, stream>>>(
        bufB, 64, W4, 50, b3, bufA, 64, N_NODES);
    zero(bufB, N_NODES * 50);
    zero(sums, 100 + N_GRAPHS * 50);        // sums + sumsq + pool (contiguous)
    scatter_add_50<<<(N_EDGES * 50 + 255) / 256, 256, 0, stream>>>(
        bufA, esrc, edst, bufB);

    // ---- BatchNorm stats + apply + global_add_pool ---------------------
    bn_stats<<<1024, 256, 0, stream>>>(bufB, sums, sumsq);
    bn_pool<<<(N_NODES * 50 + 255) / 256, 256, 0, stream>>>(
        bufB, batch, sums, sumsq, bn_g, bn_b, pool);

    // ---- MLP head + log_softmax ----------------------------------------
    head_kernel<<<1, 256, 0, stream>>>(pool, fc1W, fc1b, fc2W, fc2b, fc3W, fc3b,
                                       (float*)d_out);
}